// H_layer_85512798863503
// MI455X (gfx1250) — compile-verified
//
#include <hip/hip_runtime.h>
#include <math.h>

// Problem constants (match reference)
#define NND  50000
#define NE   800000
#define DIN  128
#define NHEAD 4
#define HD   16
#define DOUT 64   // NHEAD*HD

typedef float v2f __attribute__((ext_vector_type(2)));
typedef float v8f __attribute__((ext_vector_type(8)));

// ---- order-preserving float <-> uint mapping for atomicMax on floats ----
__device__ __forceinline__ unsigned ord_key(float f) {
  unsigned u = __float_as_uint(f);
  return (u & 0x80000000u) ? ~u : (u | 0x80000000u);
}
__device__ __forceinline__ float ord_decode(unsigned u) {
  unsigned b = (u & 0x80000000u) ? (u ^ 0x80000000u) : ~u;
  return __uint_as_float(b);
}
#define ORD_NEG_INF 0x007FFFFFu   // ord_key(-inf)

// ---------------------------------------------------------------------------
// GEMM helper: one wave computes a 16x64 output tile = (x_tile @ W) + b
// via V_WMMA_F32_16X16X4_F32.  Distinct __restrict__ pointers per call keep
// W/b/o in the global address space (global_load, not flat_load).
// ---------------------------------------------------------------------------
__device__ __forceinline__
void gemm_tile_16x64(const float* __restrict__ xt,   // LDS 16x128 tile
                     const float* __restrict__ W,    // [128,64] row-major
                     const float* __restrict__ b,    // [64]
                     float* __restrict__ o,          // [N,64]
                     int base, int r, int koff, int mofs) {
  for (int nt = 0; nt < 4; ++nt) {            // 4 x (16x16) tiles = 64 cols
    const int col = nt * 16 + r;
    v8f acc;
    const float bv = b[col];
#pragma unroll
    for (int i = 0; i < 8; ++i) acc[i] = bv;  // fold bias into C
#pragma unroll
    for (int k = 0; k < DIN / 4; ++k) {       // 32 K-steps of 4
      v2f A, B;
      A.x = xt[r * DIN + k * 4 + koff];
      A.y = xt[r * DIN + k * 4 + koff + 1];
      B.x = W[(k * 4 + koff)     * DOUT + col];
      B.y = W[(k * 4 + koff + 1) * DOUT + col];
      acc = __builtin_amdgcn_wmma_f32_16x16x4_f32(false, A, false, B,
                                                  (short)0, acc, false, false);
    }
#pragma unroll
    for (int i = 0; i < 8; ++i)
      o[(base + mofs + i) * DOUT + col] = acc[i];
  }
}

// ---------------------------------------------------------------------------
// Kernel 1: per-16-node-tile GEMMs
//   as = x@Ws + bs ; ad = x@Wd + bd ; h = x@Wl + bl
// ---------------------------------------------------------------------------
__global__ __launch_bounds__(32)
void node_gemm_kernel(const float* __restrict__ x,
                      const float* __restrict__ Ws, const float* __restrict__ bs,
                      const float* __restrict__ Wd, const float* __restrict__ bd,
                      const float* __restrict__ Wl, const float* __restrict__ bl,
                      float* __restrict__ as_, float* __restrict__ ad_,
                      float* __restrict__ h_out) {
  __shared__ float xt[16 * DIN];
  const int tile = blockIdx.x;            // N/16 = 3125 tiles exactly
  const int lane = threadIdx.x;           // 0..31 (wave32)
  const int base = tile * 16;

  // stage 16x128 tile of x into LDS (coalesced b128-friendly pattern)
  {
    const float4* xin = (const float4*)(x + (size_t)base * DIN);
    float4* xts = (float4*)xt;
    for (int i = lane; i < 16 * DIN / 4; i += 32) xts[i] = xin[i];
  }
  __syncthreads();

  const int r    = lane & 15;             // A row / B,C,D col within tile
  const int koff = (lane >> 4) * 2;       // lanes 0-15: K 0,1 ; 16-31: K 2,3
  const int mofs = (lane >> 4) * 8;       // C/D: VGPR i -> M=i (lo) / i+8 (hi)

  gemm_tile_16x64(xt, Ws, bs, as_,   base, r, koff, mofs);
  gemm_tile_16x64(xt, Wd, bd, ad_,   base, r, koff, mofs);
  gemm_tile_16x64(xt, Wl, bl, h_out, base, r, koff, mofs);
}

// ---------------------------------------------------------------------------
// Kernel 2: per-node attention partials  ps[n,h]=h[n,h,:]·Wa[0:16],
//                                        pd[n,h]=h[n,h,:]·Wa[16:32]
// ---------------------------------------------------------------------------
__global__ void pspd_kernel(const float* __restrict__ h,
                            const float* __restrict__ Wa,
                            float* __restrict__ ps, float* __restrict__ pd) {
  int n = blockIdx.x * blockDim.x + threadIdx.x;
  if (n >= NND) return;
  for (int hh = 0; hh < NHEAD; ++hh) {
    float s = 0.f, d = 0.f;
#pragma unroll
    for (int j = 0; j < HD; ++j) {
      float v = h[n * DOUT + hh * HD + j];
      s += v * Wa[j];
      d += v * Wa[HD + j];
    }
    ps[n * NHEAD + hh] = s;
    pd[n * NHEAD + hh] = d;
  }
}

// ---------------------------------------------------------------------------
// Init kernel: zero accumulators (d_out is poisoned, not re-zeroed, per call)
// ---------------------------------------------------------------------------
__global__ void init_kernel(float* __restrict__ out_accum,     // 2*N*64 floats
                            float* __restrict__ ssum,
                            float* __restrict__ deg,
                            unsigned* __restrict__ mord) {
  int i = blockIdx.x * blockDim.x + threadIdx.x;
  if (i < 2 * NND * DOUT) out_accum[i] = 0.f;
  if (i < NND * NHEAD) { ssum[i] = 0.f; mord[i] = ORD_NEG_INF; }
  if (i < NND) deg[i] = 0.f;
}

// ---------------------------------------------------------------------------
// Edge pass A (one wave per edge, 2 contiguous channels per lane, float2):
//   er = tanh(as[src]+ad[dst]); atomicAdd er -> edge_s;
//   a[e,head] = ps[src]+pd[dst]+er·Wa[32:48]+ba, leakyrelu;
//   atomicMax ordered-max; atomicAdd deg.
// ---------------------------------------------------------------------------
__global__ __launch_bounds__(256)
void edgeA_kernel(const int* __restrict__ src, const int* __restrict__ dst,
                  const float2* __restrict__ as2, const float2* __restrict__ ad2,
                  const float* __restrict__ ps, const float* __restrict__ pd,
                  const float* __restrict__ Wa, const float* __restrict__ ba,
                  float* __restrict__ a_ws, unsigned* __restrict__ mord,
                  float* __restrict__ esum, float* __restrict__ deg) {
  int e    = blockIdx.x * 8 + (threadIdx.x >> 5);   // 8 edges / block
  int lane = threadIdx.x & 31;
  if (e >= NE) return;                              // wave-uniform
  int s = src[e], d = dst[e];
  int c0 = lane * 2, c1 = c0 + 1;

  float2 va = as2[s * (DOUT / 2) + lane];           // global_load_b64
  float2 vb = ad2[d * (DOUT / 2) + lane];
  float er0 = tanhf(va.x + vb.x);
  float er1 = tanhf(va.y + vb.y);
  atomicAdd(&esum[d * DOUT + c0], er0);
  atomicAdd(&esum[d * DOUT + c1], er1);

  // per-head dot with Wa[32:48]: 8 lanes per head, xor-reduce
  float t = er0 * Wa[2 * HD + (c0 & 15)] + er1 * Wa[2 * HD + (c1 & 15)];
  t += __shfl_xor(t, 1, 32);
  t += __shfl_xor(t, 2, 32);
  t += __shfl_xor(t, 4, 32);
  int head = lane >> 3;
  if ((lane & 7) == 0) {
    float a = ps[s * NHEAD + head] + pd[d * NHEAD + head] + t + ba[0];
    a = (a >= 0.f) ? a : 0.01f * a;                 // LeakyReLU(0.01)
    a_ws[e * NHEAD + head] = a;
    atomicMax(&mord[d * NHEAD + head], ord_key(a));
  }
  if (lane == 0) atomicAdd(&deg[d], 1.0f);
}

// ---------------------------------------------------------------------------
// Edge pass B: w = exp(a - m[dst]); atomicAdd ssum; atomicAdd w*h[src] -> out
// ---------------------------------------------------------------------------
__global__ __launch_bounds__(256)
void edgeB_kernel(const int* __restrict__ src, const int* __restrict__ dst,
                  const float2* __restrict__ h2,
                  const float* __restrict__ a_ws,
                  const unsigned* __restrict__ mord,
                  float* __restrict__ ssum, float* __restrict__ num) {
  int e    = blockIdx.x * 8 + (threadIdx.x >> 5);
  int lane = threadIdx.x & 31;
  if (e >= NE) return;
  int s = src[e], d = dst[e];
  int head = lane >> 3;
  float a = a_ws[e * NHEAD + head];
  float m = ord_decode(mord[d * NHEAD + head]);     // finite: this edge signaled
  float w = expf(a - m);
  if ((lane & 7) == 0) atomicAdd(&ssum[d * NHEAD + head], w);
  float2 hv = h2[s * (DOUT / 2) + lane];            // global_load_b64
  int c0 = lane * 2;
  atomicAdd(&num[d * DOUT + c0],     w * hv.x);
  atomicAdd(&num[d * DOUT + c0 + 1], w * hv.y);
}

// ---------------------------------------------------------------------------
// Final: out /= max(ssum,1e-38) ; edge_s /= max(deg,1)
// ---------------------------------------------------------------------------
__global__ void final_kernel(const float* __restrict__ ssum,
                             const float* __restrict__ deg,
                             float* __restrict__ edge_s,
                             float* __restrict__ out) {
  int i = blockIdx.x * blockDim.x + threadIdx.x;
  if (i >= NND * DOUT) return;
  int n = i / DOUT, c = i % DOUT;
  out[i]    = out[i]    / fmaxf(ssum[n * NHEAD + (c >> 4)], 1e-38f);
  edge_s[i] = edge_s[i] / fmaxf(deg[n], 1.0f);
}

// ---------------------------------------------------------------------------
extern "C" void kernel_launch(void* const* d_in, const int* in_sizes, int n_in,
                              void* d_out, int out_size, void* d_ws, size_t ws_size,
                              hipStream_t stream) {
  const float* x   = (const float*)d_in[0];
  const int*   src = (const int*)d_in[1];
  const int*   dst = (const int*)d_in[2];
  const float* Ws  = (const float*)d_in[3];
  const float* bs  = (const float*)d_in[4];
  const float* Wd  = (const float*)d_in[5];
  const float* bd  = (const float*)d_in[6];
  const float* Wl  = (const float*)d_in[7];
  const float* bl  = (const float*)d_in[8];
  const float* Wa  = (const float*)d_in[9];
  const float* ba  = (const float*)d_in[10];

  // outputs concatenated: edge_s[N,64], out[N,64], h[N,64]
  float* edge_s = (float*)d_out;
  float* outp   = edge_s + (size_t)NND * DOUT;
  float* h      = outp   + (size_t)NND * DOUT;

  // workspace carve-up (~42 MB, all node state L2-resident on 192MB L2)
  float* ws   = (float*)d_ws;
  float* as_  = ws;  ws += (size_t)NND * DOUT;      // 12.8 MB
  float* ad_  = ws;  ws += (size_t)NND * DOUT;      // 12.8 MB
  float* ps   = ws;  ws += (size_t)NND * NHEAD;
  float* pd   = ws;  ws += (size_t)NND * NHEAD;
  float* a_ws = ws;  ws += (size_t)NE  * NHEAD;     // 12.8 MB
  unsigned* mord = (unsigned*)ws; ws += (size_t)NND * NHEAD;
  float* ssum = ws;  ws += (size_t)NND * NHEAD;
  float* deg  = ws;  ws += (size_t)NND;

  init_kernel<<<(2 * NND * DOUT + 255) / 256, 256, 0, stream>>>(edge_s, ssum, deg, mord);
  node_gemm_kernel<<<NND / 16, 32, 0, stream>>>(x, Ws, bs, Wd, bd, Wl, bl, as_, ad_, h);
  pspd_kernel<<<(NND + 255) / 256, 256, 0, stream>>>(h, Wa, ps, pd);
  edgeA_kernel<<<NE / 8, 256, 0, stream>>>(src, dst, (const float2*)as_,
                                           (const float2*)ad_, ps, pd, Wa, ba,
                                           a_ws, mord, edge_s, deg);
  edgeB_kernel<<<NE / 8, 256, 0, stream>>>(src, dst, (const float2*)h, a_ws,
                                           mord, ssum, outp);
  final_kernel<<<(NND * DOUT + 255) / 256, 256, 0, stream>>>(ssum, deg, edge_s, outp);
}